// LateralInhibition_86715389706294
// MI455X (gfx1250) — compile-verified
//
#include <hip/hip_runtime.h>
#include <math.h>

#define N_SIDE  512
#define LWIN    7
#define PAD     3
#define TILE    32
#define PATCH   (TILE + 2 * PAD)   /* 38 */
#define PATCH_W PATCH              /* TDM writes rows contiguously -> stride 38 */
#define AW      0.3f
#define BW      0.7f

#if defined(__has_builtin)
#if __has_builtin(__builtin_amdgcn_global_load_async_to_lds_b32)
#define HAVE_ASYNC_LDS 1
#endif
#if __has_builtin(__builtin_amdgcn_s_wait_asynccnt)
#define HAVE_WAIT_ASYNC_BUILTIN 1
#endif
#if __has_builtin(__builtin_amdgcn_tensor_load_to_lds) && __has_builtin(__builtin_amdgcn_s_wait_tensorcnt)
#define HAVE_TDM 1
#endif
#endif

typedef __attribute__((address_space(1))) int g_as_int;
typedef __attribute__((address_space(3))) int l_as_int;

#if defined(HAVE_TDM)
typedef unsigned int u32x4 __attribute__((ext_vector_type(4)));
typedef int          i32x8 __attribute__((ext_vector_type(8)));
typedef int          i32x4 __attribute__((ext_vector_type(4)));
#endif

__device__ __forceinline__ void wait_async_then_barrier() {
#if defined(HAVE_ASYNC_LDS)
#if defined(HAVE_WAIT_ASYNC_BUILTIN)
    __builtin_amdgcn_s_wait_asynccnt(0);
#else
    asm volatile("s_wait_asynccnt 0" ::: "memory");
#endif
#endif
    __syncthreads();
}

// Guarded per-lane staging (border tiles / fallback): async global->LDS copy
// for in-bounds cells, DS zero-fill for the zero-padding halo.
__device__ __forceinline__ void stage_patch_guarded(
    float* patch, const float* xb, int r0, int c0, int tid)
{
    for (int k = tid; k < PATCH * PATCH_W; k += 256) {
        const int i  = k / PATCH_W;
        const int j  = k - i * PATCH_W;
        const int gr = r0 - PAD + i;
        const int gc = c0 - PAD + j;
        const bool ok = (gr >= 0) & (gr < N_SIDE) & (gc >= 0) & (gc < N_SIDE);
        if (ok) {
#if defined(HAVE_ASYNC_LDS)
            __builtin_amdgcn_global_load_async_to_lds_b32(
                (g_as_int*)(xb + gr * N_SIDE + gc),
                (l_as_int*)&patch[k], 0, 0);
#else
            patch[k] = xb[gr * N_SIDE + gc];
#endif
        } else {
            patch[k] = 0.0f;
        }
    }
}

// ---------------------------------------------------------------------------
// K1: fused 7x7 stencil over one 32x32 tile per block (block = 32x8 = 256 thr,
// 8 wave32s). Interior tiles: one TDM tensor_load_to_lds stages the whole
// 38x38 halo patch (DMA engine does address gen). Border tiles: per-lane
// async global->LDS loads + zero fill. Emits avg term, diff term, and a
// per-block partial sum of mask_pre^2 (fixed-order reduction, deterministic).
// ---------------------------------------------------------------------------
__global__ __launch_bounds__(256) void li_stencil(
    const float* __restrict__ x, const float* __restrict__ kern,
    float* __restrict__ avg_out, float* __restrict__ diff_out,
    float* __restrict__ partials)
{
    __shared__ float patch[PATCH * PATCH_W];
    __shared__ float kw[LWIN * LWIN];
    __shared__ float wsum[8];

    const int tx  = threadIdx.x;              // 0..31
    const int ty  = threadIdx.y;              // 0..7
    const int tid = ty * 32 + tx;
    const int c0  = blockIdx.x * TILE;
    const int r0  = blockIdx.y * TILE;
    const int b   = blockIdx.z;
    const float* xb = x + (size_t)b * N_SIDE * N_SIDE;

    if (tid < LWIN * LWIN) kw[tid] = kern[tid];

#if defined(HAVE_TDM)
    const bool interior = (blockIdx.x > 0) & (blockIdx.x + 1 < gridDim.x) &
                          (blockIdx.y > 0) & (blockIdx.y + 1 < gridDim.y);
    if (interior) {
        if (tid == 0) {
            const unsigned long long gaddr =
                (unsigned long long)(const void*)(xb + (r0 - PAD) * N_SIDE + (c0 - PAD));
            const unsigned int lds_addr =
                (unsigned int)(unsigned long long)(l_as_int*)&patch[0];
            const unsigned int TDIM = 1u << 20;   // >> any in-tile coordinate
            // D# group0: count=1 | lds_addr | global_addr[56:0] | type=2
            u32x4 g0 = { 1u,
                         lds_addr,
                         (unsigned int)(gaddr & 0xFFFFFFFFull),
                         (((unsigned int)(gaddr >> 32)) & 0x01FFFFFFu) | (2u << 30) };
            // D# group1: data_size=4B; tensor dims huge (interior: no OOB);
            // tile 38x38; tensor_dim0_stride = 512 floats.
            i32x8 g1 = { 0x00020000,
                         (int)((TDIM & 0xFFFFu) << 16),
                         (int)((TDIM >> 16) | ((TDIM & 0xFFFFu) << 16)),
                         (int)((TDIM >> 16) | ((unsigned)PATCH << 16)),
                         PATCH,
                         N_SIDE,
                         0, 0 };
            i32x4 g2 = { 0, 0, 0, 0 };
            i32x4 g3 = { 0, 0, 0, 0 };
            i32x8 g4 = { 0, 0, 0, 0, 0, 0, 0, 0 };
            __builtin_amdgcn_tensor_load_to_lds(g0, g1, g2, g3, g4, 0);
        }
        __builtin_amdgcn_s_wait_tensorcnt(0);
    } else {
        stage_patch_guarded(patch, xb, r0, c0, tid);
    }
#else
    stage_patch_guarded(patch, xb, r0, c0, tid);
#endif
    wait_async_then_barrier();

    float s2 = 0.0f;
    #pragma unroll
    for (int rr = 0; rr < 4; ++rr) {
        const int row    = ty + rr * 8;   // 0..31
        const float center = patch[(row + PAD) * PATCH_W + (tx + PAD)];
        float mean = 0.0f;
        float diff = 0.0f;
        #pragma unroll
        for (int di = 0; di < LWIN; ++di) {
            const float* prow = &patch[(row + di) * PATCH_W + tx];
            #pragma unroll
            for (int dj = 0; dj < LWIN; ++dj) {
                const float s = prow[dj];
                mean += s;
                diff = fmaf(kw[di * LWIN + dj], fmaxf(s - center, 0.0f), diff);
            }
        }
        const float avg = __expf(mean * (-1.0f / (LWIN * LWIN)));
        const size_t idx = ((size_t)b * N_SIDE + (size_t)(r0 + row)) * N_SIDE + (c0 + tx);
        avg_out[idx]  = avg;
        diff_out[idx] = diff;
        const float m = AW * avg + BW * diff;
        s2 = fmaf(m, m, s2);
    }

    // wave32 butterfly reduce, then cross-wave reduce (fixed order: deterministic)
    #pragma unroll
    for (int off = 16; off > 0; off >>= 1) s2 += __shfl_xor(s2, off, 32);
    if (tx == 0) wsum[ty] = s2;
    __syncthreads();
    if (tid == 0) {
        float t = 0.0f;
        #pragma unroll
        for (int i = 0; i < 8; ++i) t += wsum[i];
        partials[((size_t)blockIdx.z * gridDim.y + blockIdx.y) * gridDim.x + blockIdx.x] = t;
    }
}

// ---------------------------------------------------------------------------
// K2: deterministic single-block reduction of block partials -> 1/sqrt(total)
// ---------------------------------------------------------------------------
__global__ __launch_bounds__(256) void li_reduce(
    const float* __restrict__ partials, int n, float* __restrict__ inv_out)
{
    __shared__ float ws[8];
    float s = 0.0f;
    for (int i = threadIdx.x; i < n; i += 256) s += partials[i];
    #pragma unroll
    for (int off = 16; off > 0; off >>= 1) s += __shfl_xor(s, off, 32);
    if ((threadIdx.x & 31) == 0) ws[threadIdx.x >> 5] = s;
    __syncthreads();
    if (threadIdx.x == 0) {
        float t = 0.0f;
        #pragma unroll
        for (int i = 0; i < 8; ++i) t += ws[i];
        *inv_out = rsqrtf(t);
    }
}

// ---------------------------------------------------------------------------
// K3: out = (x > mask_pre * inv_norm) ? x : 0, float4-vectorized.
// avg/diff were just written -> mostly resident in the 192MB L2.
// ---------------------------------------------------------------------------
__global__ __launch_bounds__(256) void li_final(
    const float* __restrict__ x, const float* __restrict__ avg_t,
    const float* __restrict__ diff_t, const float* __restrict__ inv_p,
    float* __restrict__ out, int n4)
{
    const float inv = *inv_p;
    const int i = blockIdx.x * 256 + threadIdx.x;
    if (i < n4) {
        const float4 xv = ((const float4*)x)[i];
        const float4 av = ((const float4*)avg_t)[i];
        const float4 dv = ((const float4*)diff_t)[i];
        float4 o;
        o.x = (xv.x > (AW * av.x + BW * dv.x) * inv) ? xv.x : 0.0f;
        o.y = (xv.y > (AW * av.y + BW * dv.y) * inv) ? xv.y : 0.0f;
        o.z = (xv.z > (AW * av.z + BW * dv.z) * inv) ? xv.z : 0.0f;
        o.w = (xv.w > (AW * av.w + BW * dv.w) * inv) ? xv.w : 0.0f;
        ((float4*)out)[i] = o;
    }
}

extern "C" void kernel_launch(void* const* d_in, const int* in_sizes, int n_in,
                              void* d_out, int out_size, void* d_ws, size_t ws_size,
                              hipStream_t stream) {
    const float* x    = (const float*)d_in[0];
    const float* kern = (const float*)d_in[1];
    float* out = (float*)d_out;

    const int N     = in_sizes[0];                 // 64*512*512 = 16,777,216
    const int batch = N / (N_SIDE * N_SIDE);       // 64
    float* avg_out  = out + N;                     // tuple order: out, avg, diff
    float* diff_out = out + 2 * (size_t)N;

    const int tiles   = N_SIDE / TILE;             // 16
    const int nblocks = batch * tiles * tiles;     // 16384 partials
    float* partials = (float*)d_ws;
    float* inv_p    = partials + nblocks;

    dim3 gA(tiles, tiles, batch), bA(32, 8);
    li_stencil<<<gA, bA, 0, stream>>>(x, kern, avg_out, diff_out, partials);
    li_reduce<<<1, 256, 0, stream>>>(partials, nblocks, inv_p);

    const int n4 = N / 4;
    li_final<<<(n4 + 255) / 256, 256, 0, stream>>>(x, avg_out, diff_out, inv_p, out, n4);
}